// NonParaGINConv_34668976013867
// MI455X (gfx1250) — compile-verified
//
#include <hip/hip_runtime.h>

// out = (1+eps)*feat + segment_sum(feat[src], dst),  eps = 0
// feat: [N=100000, D=128] f32, src/dst: [E=1600000] i32, out: [N, D] f32
//
// Strategy (MI455X):
//  - feat/out (51.2 MB each) are L2-resident (192 MB L2): random gather +
//    f32 scatter-atomics both resolve in L2, not HBM. Total working set
//    (feat + out + edge lists) ~115 MB < 192 MB.
//  - Kernel A (scatter): one wave32 per 4 edges. Lane l moves a float4
//    (global_load_b128 gather) and issues 4 no-return global_atomic_add_f32.
//    Edge indices are wave-uniform -> readfirstlane -> scalar s_load_b128.
//  - Kernel B (init): out = feat (float4 streaming copy; re-init every call
//    since the harness poisons d_out). Launched BEFORE the scatter kernel.
//
// Scatter kernel placed first in the file so the disasm snippet shows its
// atomic lowering.

#define D_FEAT 128

__device__ __forceinline__ void atomic_add_f32_hw(float* p, float v) {
  // Lowers to GLOBAL_ATOMIC_ADD_F32 (no return -> STOREcnt, executed at L2)
  unsafeAtomicAdd(p, v);
}

__global__ __launch_bounds__(256) void gin_scatter_kernel(
    const float* __restrict__ feat,
    const int* __restrict__ src,
    const int* __restrict__ dst,
    float* __restrict__ out,
    int nEdges) {
  const int lane = threadIdx.x & 31;
  const int wave = (blockIdx.x * 256 + threadIdx.x) >> 5;
  // e0 is wave-uniform; force SGPR so index loads become scalar SMEM loads.
  const int e0 = __builtin_amdgcn_readfirstlane(wave << 2);
  if (e0 >= nEdges) return;

  const int col = lane * 4;  // this lane's float4 slot within the 128-wide row

  if (e0 + 4 <= nEdges) {
    // 4 edges per wave iteration: scalar-load 4 src + 4 dst indices at once.
    const int4 sq = *(const int4*)(src + e0);  // s_load_b128
    const int4 dq = *(const int4*)(dst + e0);  // s_load_b128

    // Issue all 4 row gathers (global_load_b128) before the atomic burst.
    const float4 v0 = *(const float4*)(feat + (size_t)sq.x * D_FEAT + col);
    const float4 v1 = *(const float4*)(feat + (size_t)sq.y * D_FEAT + col);
    const float4 v2 = *(const float4*)(feat + (size_t)sq.z * D_FEAT + col);
    const float4 v3 = *(const float4*)(feat + (size_t)sq.w * D_FEAT + col);

    float* o0 = out + (size_t)dq.x * D_FEAT + col;
    float* o1 = out + (size_t)dq.y * D_FEAT + col;
    float* o2 = out + (size_t)dq.z * D_FEAT + col;
    float* o3 = out + (size_t)dq.w * D_FEAT + col;

    atomic_add_f32_hw(o0 + 0, v0.x);
    atomic_add_f32_hw(o0 + 1, v0.y);
    atomic_add_f32_hw(o0 + 2, v0.z);
    atomic_add_f32_hw(o0 + 3, v0.w);

    atomic_add_f32_hw(o1 + 0, v1.x);
    atomic_add_f32_hw(o1 + 1, v1.y);
    atomic_add_f32_hw(o1 + 2, v1.z);
    atomic_add_f32_hw(o1 + 3, v1.w);

    atomic_add_f32_hw(o2 + 0, v2.x);
    atomic_add_f32_hw(o2 + 1, v2.y);
    atomic_add_f32_hw(o2 + 2, v2.z);
    atomic_add_f32_hw(o2 + 3, v2.w);

    atomic_add_f32_hw(o3 + 0, v3.x);
    atomic_add_f32_hw(o3 + 1, v3.y);
    atomic_add_f32_hw(o3 + 2, v3.z);
    atomic_add_f32_hw(o3 + 3, v3.w);
  } else {
    // Tail (E % 4 != 0 in the general case; E = 1.6M is actually a multiple
    // of 4, but keep this correct for any size).
    for (int e = e0; e < nEdges; ++e) {
      const int s = src[e];
      const int d = dst[e];
      const float4 v = *(const float4*)(feat + (size_t)s * D_FEAT + col);
      float* o = out + (size_t)d * D_FEAT + col;
      atomic_add_f32_hw(o + 0, v.x);
      atomic_add_f32_hw(o + 1, v.y);
      atomic_add_f32_hw(o + 2, v.z);
      atomic_add_f32_hw(o + 3, v.w);
    }
  }
}

__global__ __launch_bounds__(256) void gin_init_kernel(
    const float* __restrict__ feat, float* __restrict__ out, int n4) {
  int i = blockIdx.x * 256 + threadIdx.x;
  if (i < n4) {
    float4 v = ((const float4*)feat)[i];
    // (1 + eps) * feat with eps == 0
    ((float4*)out)[i] = v;
  }
}

extern "C" void kernel_launch(void* const* d_in, const int* in_sizes, int n_in,
                              void* d_out, int out_size, void* d_ws, size_t ws_size,
                              hipStream_t stream) {
  const float* feat = (const float*)d_in[0];
  const int* src = (const int*)d_in[1];
  const int* dst = (const int*)d_in[2];
  float* out = (float*)d_out;

  const int featElems = in_sizes[0];      // N * 128
  const int nEdges = in_sizes[1];         // E

  // Kernel 1: re-initialize out = feat (float4 granularity).
  const int n4 = featElems / 4;
  const int initBlocks = (n4 + 255) / 256;
  gin_init_kernel<<<initBlocks, 256, 0, stream>>>(feat, out, n4);

  // Kernel 2: one wave32 per 4 edges.
  const int numWaves = (nEdges + 3) / 4;
  const int scatterBlocks = (numWaves + 7) / 8;  // 8 waves per 256-thread block
  gin_scatter_kernel<<<scatterBlocks, 256, 0, stream>>>(feat, src, dst, out, nEdges);
}